// YOLOLoss_42872363548741
// MI455X (gfx1250) — compile-verified
//
#include <hip/hip_runtime.h>

#define BATCH 16
#define SDIM  22743            // 3*(76^2 + 38^2 + 19^2)
#define S1DIM 17328            // 76*76*3
#define CDIM  85
#define GG    5776             // 76*76

typedef __attribute__((ext_vector_type(2))) float v2f;
typedef __attribute__((ext_vector_type(8))) float v8f;

// Sum a value across all 32 lanes of a wave.
// Uses V_WMMA_F32_16X16X4_F32 with A = ones(16x4), B = [v ; 0]:
//   D[m,n] = B[0,n] + B[1,n] = v[n] + v[n+16]  (B layout: b0 lanes0-15 = row0,
//   lanes16-31 = row1), so c[0] holds the 16 half-wave partials in lanes 0-15
//   (and duplicated in 16-31). Four shfl_xor steps finish 16 -> 1.
__device__ __forceinline__ float wave_sum32(float v) {
#if defined(__gfx1250__) && __has_builtin(__builtin_amdgcn_wmma_f32_16x16x4_f32)
  v2f a; a[0] = 1.0f; a[1] = 1.0f;      // A = all-ones 16x4
  v2f b; b[0] = v;    b[1] = 0.0f;      // B rows {0,1} = lane halves, rows {2,3} = 0
  v8f c = {};
  c = __builtin_amdgcn_wmma_f32_16x16x4_f32(
      /*neg_a=*/false, a, /*neg_b=*/false, b,
      /*c_mod=*/(short)0, c, /*reuse_a=*/false, /*reuse_b=*/false);
  float w = c[0];                        // lane i: v[i%16] + v[i%16 + 16]
  w += __shfl_xor(w, 8, 32);
  w += __shfl_xor(w, 4, 32);
  w += __shfl_xor(w, 2, 32);
  w += __shfl_xor(w, 1, 32);
  return w;
#else
  for (int off = 16; off > 0; off >>= 1) v += __shfl_xor(v, off, 32);
  return v;
#endif
}

__device__ __forceinline__ float bce_f(float p, float t) {
  float lp = fmaxf(logf(p), -100.0f);
  float lq = fmaxf(logf(1.0f - p), -100.0f);
  return -(t * lp + (1.0f - t) * lq);
}

// Kernel A: per-(b,g) anchor argmax/keep mask -> bytes; also zero accumulators.
__global__ void __launch_bounds__(256) yolo_mask_kernel(
    const float* __restrict__ t, unsigned char* __restrict__ om,
    double* __restrict__ acc) {
  unsigned idx = blockIdx.x * blockDim.x + threadIdx.x;
  if (idx < 8u) acc[idx] = 0.0;          // runs before the reduction kernel
  if (idx >= BATCH * GG) return;
  unsigned b = idx / GG;
  unsigned g = idx - b * GG;
  const float AW[3] = {10.f, 16.f, 33.f};
  const float AH[3] = {13.f, 30.f, 23.f};
  unsigned base = (b * SDIM + g * 3u) * CDIM;
  float iou[3];
#pragma unroll
  for (int a = 0; a < 3; ++a) {
    float gw = t[base + (unsigned)a * CDIM + 2];
    float gh = t[base + (unsigned)a * CDIM + 3];
    float inter = fminf(AW[a], gw) * fminf(AH[a], gh);  // centers are 0
    float uni = AW[a] * AH[a] + gw * gh - inter + 1e-16f;
    iou[a] = inter / uni;
  }
  int best = 0;                           // jnp.argmax: first max (strict >)
  if (iou[1] > iou[0]) best = 1;
  if (iou[2] > iou[best]) best = 2;
  unsigned obase = b * S1DIM + g * 3u;
#pragma unroll
  for (int a = 0; a < 3; ++a)
    om[obase + (unsigned)a] = (a == best || iou[a] <= 0.7f) ? 1 : 0;
}

// Kernel B: streaming float4 reduction of all BCE terms.
__global__ void __launch_bounds__(256) yolo_loss_kernel(
    const float* __restrict__ x, const float* __restrict__ t,
    const unsigned char* __restrict__ om, double* __restrict__ acc) {
  const unsigned NQ = (BATCH * SDIM * CDIM) / 4u;   // 7,732,620 quads
  float sxy = 0.f, swh = 0.f, scls = 0.f, sobj = 0.f, scnt = 0.f;
  unsigned stride = gridDim.x * blockDim.x;
  for (unsigned q = blockIdx.x * blockDim.x + threadIdx.x; q < NQ; q += stride) {
    float4 xv = reinterpret_cast<const float4*>(x)[q];   // global_load_b128
    float4 tv = reinterpret_cast<const float4*>(t)[q];
    unsigned e0   = q * 4u;
    unsigned row0 = e0 / CDIM;
    unsigned row3 = (e0 + 3u) / CDIM;                    // rows change <= once/quad
    float m0 = (t[row0 * CDIM + 4u] > 0.f) ? 1.f : 0.f;
    unsigned b0 = row0 / SDIM, s0 = row0 - b0 * SDIM;
    float o0 = (s0 < S1DIM) ? (float)om[b0 * S1DIM + s0] : 0.f;
    float m3 = m0, o3 = o0;
    if (row3 != row0) {
      m3 = (t[row3 * CDIM + 4u] > 0.f) ? 1.f : 0.f;
      unsigned b3 = row3 / SDIM, s3 = row3 - b3 * SDIM;
      o3 = (s3 < S1DIM) ? (float)om[b3 * S1DIM + s3] : 0.f;
    }
    float xa[4] = {xv.x, xv.y, xv.z, xv.w};
    float ta[4] = {tv.x, tv.y, tv.z, tv.w};
#pragma unroll
    for (int j = 0; j < 4; ++j) {
      unsigned e   = e0 + (unsigned)j;
      unsigned row = e / CDIM;
      unsigned c   = e - row * CDIM;
      float v = bce_f(xa[j], ta[j]);
      float m = (row == row0) ? m0 : m3;
      float o = (row == row0) ? o0 : o3;
      if (c < 4u)       { if (c < 2u) sxy += m * v; else swh += m * v; }
      else if (c == 4u) { scls += m * v; scnt += m; }
      sobj += o * v;                     // o==0 outside s<S1 or masked-off
    }
  }
  // wave (WMMA + shfl) -> block (LDS) -> global (f64 atomics)
  float r[5] = {sxy, swh, scls, sobj, scnt};
#pragma unroll
  for (int i = 0; i < 5; ++i) r[i] = wave_sum32(r[i]);   // uniform: EXEC all-1s
  __shared__ float red[8][5];
  unsigned lane = threadIdx.x & 31u, wv = threadIdx.x >> 5;
  if (lane == 0u) {
#pragma unroll
    for (int i = 0; i < 5; ++i) red[wv][i] = r[i];
  }
  __syncthreads();
  if (threadIdx.x == 0u) {
#pragma unroll
    for (int i = 0; i < 5; ++i) {
      double a = 0.0;
      for (int w = 0; w < 8; ++w) a += (double)red[w][i];
      atomicAdd(&acc[i], a);             // global_atomic_add_f64
    }
  }
}

// Kernel C: combine partial sums into the scalar loss.
__global__ void yolo_fin_kernel(const double* __restrict__ acc,
                                float* __restrict__ out) {
  double cnt = (acc[4] > 1.0) ? acc[4] : 1.0;
  double inv = 1.0 / cnt;
  const double denom = (double)BATCH * (double)S1DIM * (double)CDIM;
  double loss = 0.5 * acc[0] * inv   // loss_bbox_xy: /(cnt*2)
              + 0.5 * acc[1] * inv   // loss_bbox_wh: /(cnt*2)
              + acc[2] * inv         // loss_classes: /cnt
              + acc[3] / denom;      // loss_objectness: mean over B*S1*C
  out[0] = (float)loss;
}

extern "C" void kernel_launch(void* const* d_in, const int* in_sizes, int n_in,
                              void* d_out, int out_size, void* d_ws, size_t ws_size,
                              hipStream_t stream) {
  const float* x = (const float*)d_in[0];
  const float* t = (const float*)d_in[1];
  double* acc = (double*)d_ws;                               // 8 doubles
  unsigned char* om = (unsigned char*)d_ws + 64;             // B*S1 bytes (~277 KB)

  dim3 blkA((BATCH * GG + 255) / 256);
  yolo_mask_kernel<<<blkA, 256, 0, stream>>>(t, om, acc);
  yolo_loss_kernel<<<4096, 256, 0, stream>>>(x, t, om, acc);
  yolo_fin_kernel<<<1, 1, 0, stream>>>(acc, (float*)d_out);
}